// LocalGNN_39058432590325
// MI455X (gfx1250) — compile-verified
//
#include <hip/hip_runtime.h>

typedef __attribute__((ext_vector_type(4))) float v4f;
typedef __attribute__((ext_vector_type(2))) float v2f;
typedef __attribute__((ext_vector_type(8))) float v8f;

#define NCH       32
#define T_LEN     8192
#define NWIN      1024      // T / 8
#define TILE_W    32        // pooled windows per wave-unit
#define TILE_T    256       // time samples per wave-unit
#define ROWS      40        // LDS row stride (floats); 2*40 mod 64 = 16 -> conflict-free split reads
#define NGRP      11
#define WPB       8         // waves per block
#define NBATCH    512
#define TILES_PER_BATCH (NWIN / TILE_W)   // 32

// channels that belong to any group (7, 18, 25 are dead)
#define USED_MASK 0xFDFBFF7Fu

__global__ __launch_bounds__(256)
void localgnn_fused_kernel(const float* __restrict__ x,
                           const float* __restrict__ Wm,
                           const float* __restrict__ bv,
                           float* __restrict__ out) {
  // GMASK[m]: bit c set iff channel c belongs to group m (groups 11..15 empty)
  constexpr unsigned GMASK[16] = {
    0x00000003u,  // g0 : ch 0,1
    0x0000000Cu,  // g1 : ch 2,3
    0x00000030u,  // g2 : ch 4,5
    0x00030000u,  // g3 : ch 16,17
    0x00180000u,  // g4 : ch 19,20
    0x00600000u,  // g5 : ch 21,22
    0x01800040u,  // g6 : ch 6,23,24
    0x08000300u,  // g7 : ch 8,9,27
    0x34008C00u,  // g8 : ch 10,11,15,26,28,29
    0x40001000u,  // g9 : ch 12,30
    0x80006000u,  // g10: ch 13,14,31
    0u, 0u, 0u, 0u, 0u
  };

  __shared__ float zf_all[WPB][NCH * ROWS];   // 8 * 32 * 40 * 4B = 40 KB

  const int lane = threadIdx.x & 31;
  const int wid  = threadIdx.x >> 5;
  const int unit = blockIdx.x * WPB + wid;        // 0 .. 16383
  const int bidx = unit >> 5;                     // batch index (32 tiles/batch)
  const int tile = unit & (TILES_PER_BATCH - 1);  // window-tile index

  float* zf = zf_all[wid];

  const int hi = lane >> 4;
  const int n  = lane & 15;
  // Per-lane A-matrix source bits: one mask lookup, hoisted out of the WMMA loop.
  const unsigned gmh = GMASK[n] >> (2 * hi);

  // ---------------- Phase 1: fused affine + ReLU + avgpool(8) -> LDS --------
  // Lane L handles window L of this tile: 8 consecutive time samples.
  const size_t xbase = ((size_t)bidx * NCH) * T_LEN + (size_t)tile * TILE_T;
#pragma unroll
  for (int c = 0; c < NCH; ++c) {
    if (!((USED_MASK >> c) & 1u)) {   // dead channel: contributes nothing; keep LDS clean
      zf[c * ROWS + lane] = 0.0f;
      continue;
    }
    const v4f* xp = (const v4f*)(x + xbase + (size_t)c * T_LEN) + lane * 2;
    const v4f* wp = (const v4f*)(Wm + (size_t)c * T_LEN + (size_t)tile * TILE_T) + lane * 2;
    v4f p0 = __builtin_nontemporal_load(xp);      // streamed, no reuse
    v4f p1 = __builtin_nontemporal_load(xp + 1);
    v4f q0 = wp[0];                               // W reused across batches -> keep cached
    v4f q1 = wp[1];
    const float nb = -bv[c];
    float s;
    s  = fmaxf(fmaf(p0.x, q0.x, nb), 0.0f);
    s += fmaxf(fmaf(p0.y, q0.y, nb), 0.0f);
    s += fmaxf(fmaf(p0.z, q0.z, nb), 0.0f);
    s += fmaxf(fmaf(p0.w, q0.w, nb), 0.0f);
    s += fmaxf(fmaf(p1.x, q1.x, nb), 0.0f);
    s += fmaxf(fmaf(p1.y, q1.y, nb), 0.0f);
    s += fmaxf(fmaf(p1.z, q1.z, nb), 0.0f);
    s += fmaxf(fmaf(p1.w, q1.w, nb), 0.0f);
    zf[c * ROWS + lane] = s * 0.125f;             // pooled value, window = lane
  }

  // ---------------- Phase 2: grouped mean = G(11x32) x zf(32x32) via WMMA ---
  // V_WMMA_F32_16X16X4_F32, K accumulated over 8 steps (32 channels).
  // A (16x4): lanes0-15 = rows M, VGPR0/1 = K0/K1; lanes16-31 = K2/K3.
  // B (4x16): lane -> column N (mod 16), VGPR0/1 + lane-half -> K.
  v8f acc0 = {};   // windows 0..15 of tile
  v8f acc1 = {};   // windows 16..31 of tile
#pragma unroll
  for (int k = 0; k < 8; ++k) {
    // A values for channels 4k+2*hi (+0, +1): pure VALU from the hoisted mask.
    v2f a;
    a.x = (float)((gmh >> (4 * k)) & 1u);
    a.y = (float)((gmh >> (4 * k + 1)) & 1u);
    const float* row = zf + (4 * k + 2 * hi) * ROWS;
    v2f b0, b1;
    b0.x = row[n];        b0.y = row[ROWS + n];
    b1.x = row[16 + n];   b1.y = row[ROWS + 16 + n];
    acc0 = __builtin_amdgcn_wmma_f32_16x16x4_f32(false, a, false, b0,
                                                 (short)0, acc0, false, false);
    acc1 = __builtin_amdgcn_wmma_f32_16x16x4_f32(false, a, false, b1,
                                                 (short)0, acc1, false, false);
  }

  // ---------------- Scale by 1/|group| and store ----------------------------
  // D layout: VGPR r, lanes0-15 -> (M=r, N=lane); lanes16-31 -> (M=r+8, N=lane-16)
  float* outp = out + (size_t)bidx * NGRP * NWIN + (size_t)tile * TILE_W;
#pragma unroll
  for (int r = 0; r < 8; ++r) {
    const int g = r + 8 * hi;
    if (g < NGRP) {
      const float inv = (g == 8) ? (1.0f / 6.0f)
                      : ((g == 6) || (g == 7) || (g == 10)) ? (1.0f / 3.0f)
                      : 0.5f;
      __builtin_nontemporal_store(acc0[r] * inv, outp + (size_t)g * NWIN + n);
      __builtin_nontemporal_store(acc1[r] * inv, outp + (size_t)g * NWIN + 16 + n);
    }
  }
}

extern "C" void kernel_launch(void* const* d_in, const int* in_sizes, int n_in,
                              void* d_out, int out_size, void* d_ws, size_t ws_size,
                              hipStream_t stream) {
  const float* x  = (const float*)d_in[0];   // [512, 32, 8192]
  const float* Wm = (const float*)d_in[1];   // [32, 8192]
  const float* bv = (const float*)d_in[2];   // [32, 1]
  float* out = (float*)d_out;                // [512, 11, 1024]
  (void)in_sizes; (void)n_in; (void)out_size; (void)d_ws; (void)ws_size;

  const int units  = NBATCH * TILES_PER_BATCH;   // 16384 wave-units
  const int blocks = units / WPB;                // 2048
  hipLaunchKernelGGL(localgnn_fused_kernel, dim3(blocks), dim3(256), 0, stream,
                     x, Wm, bv, out);
}